// MonotonicNN_21242908246779
// MI455X (gfx1250) — compile-verified
//
#include <hip/hip_runtime.h>
#include <math.h>

typedef __attribute__((ext_vector_type(2))) float v2f;
typedef __attribute__((ext_vector_type(8))) float v8f;

#define NB_STEPS 50
#define S_STEPS 51
#define PI_D 3.14159265358979323846

static __device__ __forceinline__ v8f wmma4(v2f a, v2f b, v8f c) {
  // D = A(16x4,f32) x B(4x16,f32) + C(16x16,f32)
  return __builtin_amdgcn_wmma_f32_16x16x4_f32(false, a, false, b, (short)0, c,
                                               false, false);
}

static __device__ __forceinline__ v8f splat8(float x) {
  v8f r;
#pragma unroll
  for (int i = 0; i < 8; ++i) r[i] = x;
  return r;
}

static __device__ __forceinline__ float relu1(float v) {
#if __has_builtin(__builtin_amdgcn_fmed3f)
  // Single v_med3_num_f32: median{v, 0, +inf} == max(v, 0) for finite v
  // (WMMA outputs here are finite). Avoids fmaxf's canonicalize+max pair.
  return __builtin_amdgcn_fmed3f(v, 0.0f, __builtin_inff());
#else
  return fmaxf(v, 0.0f);
#endif
}

static __device__ __forceinline__ v8f relu8(v8f v) {
#pragma unroll
  for (int i = 0; i < 8; ++i) v[i] = relu1(v[i]);
  return v;
}

// Store a 16x16 D-tile (n-tile t) to LDS in transposed [col][row] layout:
// each lane holds rows (half*8 .. half*8+7) of column (t*16+m) -> 2x b128.
static __device__ __forceinline__ void store_tile_tr(float* act, int t, int m,
                                                     int half, v8f z) {
  float4* p = (float4*)&act[(t * 16 + m) * 16 + half * 8];
  p[0] = make_float4(z[0], z[1], z[2], z[3]);
  p[1] = make_float4(z[4], z[5], z[6], z[7]);
}

// Load A-fragment (16x4 k-chunk kc) from transposed LDS activations.
// A layout: v0 = A[m][4kc+2*half], v1 = A[m][4kc+2*half+1].
static __device__ __forceinline__ v2f load_afrag(const float* act, int kc,
                                                 int m, int half) {
  int r0 = kc * 4 + 2 * half;
  v2f a;
  a[0] = act[r0 * 16 + m];
  a[1] = act[(r0 + 1) * 16 + m];
  return a;
}

// Compute Clenshaw-Curtis weights (double, matching the fp64 reference) and
// the scaled step positions (steps[s]+1)/2 into d_ws: [0..63]=tsc, [64..127]=ccw.
__global__ void cc_init_kernel(float* ws) {
  int i = threadIdx.x;
  if (i > NB_STEPS) return;
  double st = cos((double)i * PI_D / (double)NB_STEPS);
  double acc = 0.0;
  for (int j = 0; j <= NB_STEPS; j += 2) {  // odd j have W[j]==0
    double Wj = (j == 0) ? 1.0 : 2.0 / (1.0 - (double)j * (double)j);
    double lam;
    if (i == 0) {
      lam = 0.5;
    } else {
      lam = cos((double)j * (double)i * PI_D / (double)NB_STEPS);
      if (i == NB_STEPS) lam *= 0.5;
    }
    acc += lam * (2.0 / (double)NB_STEPS) * Wj;
  }
  ws[i] = (float)((st + 1.0) * 0.5);
  ws[64 + i] = (float)acc;
}

__global__ void __launch_bounds__(32) monnn_kernel(
    const float* __restrict__ x, const float* __restrict__ h,
    const float* __restrict__ iW1, const float* __restrict__ ib1,
    const float* __restrict__ iW2, const float* __restrict__ ib2,
    const float* __restrict__ iW3, const float* __restrict__ ib3,
    const float* __restrict__ sW1, const float* __restrict__ sb1,
    const float* __restrict__ sW2, const float* __restrict__ sb2,
    const float* __restrict__ sW3, const float* __restrict__ sb3,
    const float* __restrict__ ws, float* __restrict__ out) {
  __shared__ __align__(16) float actA[1024];   // 64 cols x 16 rows, transposed
  __shared__ __align__(16) float actB[1024];
  __shared__ __align__(16) float offs_s[16];
  __shared__ __align__(16) float scl_s[16];
  __shared__ __align__(16) float accst[16];
  __shared__ float xs[16];
  __shared__ float tsc_s[S_STEPS];
  __shared__ float ccw_s[S_STEPS];

  const int lane = threadIdx.x & 31;
  const int m = lane & 15;      // row (A) / col (B,D) index
  const int half = lane >> 4;   // lane-half selects K/M sub-block
  const int b0 = blockIdx.x * 16;

  for (int i = lane; i < S_STEPS; i += 32) {
    tsc_s[i] = ws[i];
    ccw_s[i] = ws[64 + i];
  }

  const float xm = x[b0 + m];
  float hv[7];
#pragma unroll
  for (int j = 0; j < 7; ++j) hv[j] = h[(b0 + m) * 7 + j];
  if (half == 0) xs[m] = xm;
  __syncthreads();  // single-wave workgroup: ~free (S_NOP barrier)

  // ---------------- Scaling network: s = MLP(h), 7->64->64->2 ----------------
  // A cols: K0..K6 = h0..h6, K7 = 0.
  v2f sa0, sa1;
  sa0[0] = half ? hv[2] : hv[0];
  sa0[1] = half ? hv[3] : hv[1];
  sa1[0] = half ? hv[6] : hv[4];
  sa1[1] = half ? 0.0f : hv[5];
  {
#pragma unroll
    for (int t = 0; t < 4; ++t) {
      v8f c = splat8(sb1[t * 16 + m]);
      v2f bb;
      int r0 = 2 * half;
      bb[0] = sW1[r0 * 64 + t * 16 + m];
      bb[1] = sW1[(r0 + 1) * 64 + t * 16 + m];
      c = wmma4(sa0, bb, c);
      // chunk1 rows: half0 -> {4,5}, half1 -> {6,7(pad)}. Row 7 would be OOB:
      // load a clamped valid row unconditionally, zero it with a select
      // (matching A col7 == 0), so no exec-mask divergence is generated.
      int r1 = 4 + 2 * half;            // 4 or 6
      int r2 = half ? 6 : 5;            // clamped second row (always valid)
      float bv0 = sW1[r1 * 64 + t * 16 + m];
      float bv1 = sW1[r2 * 64 + t * 16 + m];
      bb[0] = bv0;
      bb[1] = half ? 0.0f : bv1;
      c = wmma4(sa1, bb, c);
      store_tile_tr(actA, t, m, half, relu8(c));
    }
  }
  __syncthreads();
  {
    v8f z[4];
#pragma unroll
    for (int t = 0; t < 4; ++t) z[t] = splat8(sb2[t * 16 + m]);
#pragma unroll
    for (int kc = 0; kc < 16; ++kc) {
      v2f a = load_afrag(actA, kc, m, half);
      int r0 = kc * 4 + 2 * half;
#pragma unroll
      for (int t = 0; t < 4; ++t) {
        v2f bb;
        bb[0] = sW2[r0 * 64 + t * 16 + m];
        bb[1] = sW2[(r0 + 1) * 64 + t * 16 + m];
        z[t] = wmma4(a, bb, z[t]);
      }
    }
#pragma unroll
    for (int t = 0; t < 4; ++t) store_tile_tr(actB, t, m, half, relu8(z[t]));
  }
  __syncthreads();
  {
    // 64->2: put sW3 columns 0,1 into B cols 0,1 (rest zero).
    // Unconditional loads from a clamped column + value select (branch-free).
    const int colc = (m < 2) ? m : 0;
    const bool live = (m < 2);
    v8f st = splat8(live ? sb3[colc] : 0.0f);
#pragma unroll
    for (int kc = 0; kc < 16; ++kc) {
      v2f a = load_afrag(actB, kc, m, half);
      int r0 = kc * 4 + 2 * half;
      float bv0 = sW3[r0 * 2 + colc];
      float bv1 = sW3[(r0 + 1) * 2 + colc];
      v2f bb;
      bb[0] = live ? bv0 : 0.0f;
      bb[1] = live ? bv1 : 0.0f;
      st = wmma4(a, bb, st);
    }
    if (m == 0) {
      *(float4*)&offs_s[half * 8] = make_float4(st[0], st[1], st[2], st[3]);
      *(float4*)&offs_s[half * 8 + 4] = make_float4(st[4], st[5], st[6], st[7]);
    }
    if (m == 1) {
      *(float4*)&scl_s[half * 8] = make_float4(st[0], st[1], st[2], st[3]);
      *(float4*)&scl_s[half * 8 + 4] = make_float4(st[4], st[5], st[6], st[7]);
    }
  }
  __syncthreads();

  // ------------- Integral network weights, hoisted to registers -------------
  v2f w1f[2][4];
#pragma unroll
  for (int c = 0; c < 2; ++c) {
#pragma unroll
    for (int t = 0; t < 4; ++t) {
      int r0 = c * 4 + 2 * half;
      w1f[c][t][0] = iW1[r0 * 64 + t * 16 + m];
      w1f[c][t][1] = iW1[(r0 + 1) * 64 + t * 16 + m];
    }
  }
  v2f w2f[16][4];
#pragma unroll
  for (int kc = 0; kc < 16; ++kc) {
    int r0 = kc * 4 + 2 * half;
#pragma unroll
    for (int t = 0; t < 4; ++t) {
      w2f[kc][t][0] = iW2[r0 * 64 + t * 16 + m];
      w2f[kc][t][1] = iW2[(r0 + 1) * 64 + t * 16 + m];
    }
  }
  v2f w3f[16];  // iW3 broadcast across all 16 B-columns
#pragma unroll
  for (int kc = 0; kc < 16; ++kc) {
    int r0 = kc * 4 + 2 * half;
    w3f[kc][0] = iW3[r0];
    w3f[kc][1] = iW3[r0 + 1];
  }
  float ib1v[4], ib2v[4];
#pragma unroll
  for (int t = 0; t < 4; ++t) {
    ib1v[t] = ib1[t * 16 + m];
    ib2v[t] = ib2[t * 16 + m];
  }
  const float ib3s = ib3[0];

  // A cols for integral net: K0 = X_s (changes per step), K1..K7 = h0..h6.
  v2f a1;
  a1[0] = half ? hv[5] : hv[3];
  a1[1] = half ? hv[6] : hv[4];
  const float a0_h1 = hv[1];                  // half1 K2
  const float a0y = half ? hv[2] : hv[0];     // K1 / K3

  v8f acc = splat8(0.0f);

  for (int s = 0; s < S_STEPS; ++s) {
    const float ts = tsc_s[s];
    v2f a0;
    a0[0] = half ? a0_h1 : xm * ts;  // only this element varies with s
    a0[1] = a0y;
    // L1: 8->64
#pragma unroll
    for (int t = 0; t < 4; ++t) {
      v8f c = splat8(ib1v[t]);
      c = wmma4(a0, w1f[0][t], c);
      c = wmma4(a1, w1f[1][t], c);
      store_tile_tr(actA, t, m, half, relu8(c));
    }
    __syncthreads();
    // L2: 64->64 (weights register-resident, reused 51x)
    v8f z[4];
#pragma unroll
    for (int t = 0; t < 4; ++t) z[t] = splat8(ib2v[t]);
#pragma unroll
    for (int kc = 0; kc < 16; ++kc) {
      v2f a = load_afrag(actA, kc, m, half);
#pragma unroll
      for (int t = 0; t < 4; ++t) z[t] = wmma4(a, w2f[kc][t], z[t]);
    }
#pragma unroll
    for (int t = 0; t < 4; ++t) store_tile_tr(actB, t, m, half, relu8(z[t]));
    __syncthreads();
    // L3: 64->1, iW3 broadcast so every lane holds z3 of its 8 rows
    v8f z3 = splat8(ib3s);
#pragma unroll
    for (int kc = 0; kc < 16; ++kc) {
      v2f a = load_afrag(actB, kc, m, half);
      z3 = wmma4(a, w3f[kc], z3);
    }
    const float cw = ccw_s[s];
#pragma unroll
    for (int r = 0; r < 8; ++r) {
      float v = z3[r];
      // elu(v)+1: argument of exp is <= 0 -> hardware v_exp_f32 path is
      // accurate to a few ulp there; the 51-term quadrature averages the error.
      float zz = (v > 0.0f) ? (v + 1.0f) : __expf(v);
      acc[r] = fmaf(cw, zz, acc[r]);
    }
  }

  if (m == 0) {  // lanes 0/16 hold rows half*8 .. half*8+7
    *(float4*)&accst[half * 8] = make_float4(acc[0], acc[1], acc[2], acc[3]);
    *(float4*)&accst[half * 8 + 4] = make_float4(acc[4], acc[5], acc[6], acc[7]);
  }
  __syncthreads();
  if (half == 0) {
    float integral = accst[m] * xs[m] * 0.5f;
    out[b0 + m] = expf(scl_s[m]) * integral + offs_s[m];
  }
}

extern "C" void kernel_launch(void* const* d_in, const int* in_sizes, int n_in,
                              void* d_out, int out_size, void* d_ws,
                              size_t ws_size, hipStream_t stream) {
  const float* x = (const float*)d_in[0];
  const float* h = (const float*)d_in[1];
  const float* iW1 = (const float*)d_in[2];
  const float* ib1 = (const float*)d_in[3];
  const float* iW2 = (const float*)d_in[4];
  const float* ib2 = (const float*)d_in[5];
  const float* iW3 = (const float*)d_in[6];
  const float* ib3 = (const float*)d_in[7];
  const float* sW1 = (const float*)d_in[8];
  const float* sb1 = (const float*)d_in[9];
  const float* sW2 = (const float*)d_in[10];
  const float* sb2 = (const float*)d_in[11];
  const float* sW3 = (const float*)d_in[12];
  const float* sb3 = (const float*)d_in[13];
  float* ws = (float*)d_ws;

  cc_init_kernel<<<1, 64, 0, stream>>>(ws);

  const int B = in_sizes[0];  // 32768, divisible by 16
  monnn_kernel<<<B / 16, 32, 0, stream>>>(x, h, iW1, ib1, iW2, ib2, iW3, ib3,
                                          sW1, sb1, sW2, sb2, sW3, sb3, ws,
                                          (float*)d_out);
}